// ConvAttention_43894565765370
// MI455X (gfx1250) — compile-verified
//
#include <hip/hip_runtime.h>
#include <math.h>

// Problem constants (from reference setup_inputs)
#define BATCH 8
#define HH 56
#define WW 56
#define CH 64              // channels (= contraction dim for attention)
#define NPIX (HH * WW)     // 3136
#define ED 64              // EMBED_DIM
#define QTILES (NPIX / 16) // 196 key/query tiles per batch
#define KGROUPS (QTILES / 4) // 49 groups of 4 key-tiles (64 keys each)
#define NSPLIT 4           // flash-decoding key splits per query tile
#define NTILES (BATCH * QTILES) // 1568 query tiles
#define WAVES_PER_BLOCK 4

typedef float v2f __attribute__((ext_vector_type(2)));
typedef float v8f __attribute__((ext_vector_type(8)));

// DPP16 lane butterfly within 16-lane groups (wave32: lanes 0-15 / 16-31).
// masks {^1(quad), ^2(quad), ^7(row_half_mirror), ^15(row_mirror)} span all 16.
#define DPPF(v, ctrl)                                                        \
    __int_as_float(__builtin_amdgcn_update_dpp(                              \
        __float_as_int(v), __float_as_int(v), (ctrl), 0xF, 0xF, true))

static __device__ __forceinline__ float redmax16(float v) {
    v = fmaxf(v, DPPF(v, 0xB1));  // quad_perm [1,0,3,2]  (xor 1)
    v = fmaxf(v, DPPF(v, 0x4E));  // quad_perm [2,3,0,1]  (xor 2)
    v = fmaxf(v, DPPF(v, 0x141)); // row_half_mirror      (xor 7)
    v = fmaxf(v, DPPF(v, 0x140)); // row_mirror           (xor 15)
    return v;
}
static __device__ __forceinline__ float redsum16(float v) {
    v += DPPF(v, 0xB1);
    v += DPPF(v, 0x4E);
    v += DPPF(v, 0x141);
    v += DPPF(v, 0x140);
    return v;
}

// -------------------------------------------------------------------------
// Kernel 1: fused depthwise 3x3 conv -> Q,K row-major [B,N,C], V transposed
// [B,C,N] (so the P·V WMMA B-operand is a contiguous float2 per lane).
// -------------------------------------------------------------------------
__global__ __launch_bounds__(64) void qkv_dwconv(
    const float* __restrict__ x,
    const float* __restrict__ wq, const float* __restrict__ bq,
    const float* __restrict__ wk, const float* __restrict__ bk,
    const float* __restrict__ wv, const float* __restrict__ bv,
    float* __restrict__ q, float* __restrict__ k, float* __restrict__ vt)
{
    const int c   = threadIdx.x;          // 0..63 channel
    const int pix = blockIdx.x;           // b*NPIX + n
    const int b   = pix / NPIX;
    const int n   = pix - b * NPIX;
    const int h   = n / WW;
    const int w   = n - h * WW;

    float aq = 0.f, ak = 0.f, av = 0.f;
#pragma unroll
    for (int dy = -1; dy <= 1; ++dy) {
#pragma unroll
        for (int dx = -1; dx <= 1; ++dx) {
            int hh = h + dy, ww = w + dx;
            if (hh >= 0 && hh < HH && ww >= 0 && ww < WW) {
                float xv = x[(((size_t)b * HH + hh) * WW + ww) * CH + c];
                int wi = ((dy + 1) * 3 + (dx + 1)) * CH + c;
                aq = fmaf(xv, wq[wi], aq);
                ak = fmaf(xv, wk[wi], ak);
                av = fmaf(xv, wv[wi], av);
            }
        }
    }
    q[(size_t)pix * CH + c]             = aq + bq[c];
    k[(size_t)pix * CH + c]             = ak + bk[c];
    vt[((size_t)b * CH + c) * NPIX + n] = av + bv[c]; // transposed
}

// -------------------------------------------------------------------------
// Kernel 2: flash attention partials (online softmax, 64 keys per update).
// One wave32 per (query tile, key split).  f32 WMMA 16x16x4 throughout.
// Writes unnormalized acc (D layout) + per-row (m, l) for each split.
// -------------------------------------------------------------------------
__global__ __launch_bounds__(32 * WAVES_PER_BLOCK) void flash_attn_split(
    const float* __restrict__ Q, const float* __restrict__ K,
    const float* __restrict__ Vt,
    float* __restrict__ partAcc,  // [NTILES][NSPLIT][16][64]
    float* __restrict__ partML)   // [NTILES][NSPLIT][32] : m[16], l[16]
{
    // per-wave LDS scratch: 16x64 bounce at row stride 66 -> 1054 floats
    __shared__ float lds[WAVES_PER_BLOCK][1056];

    const int wave = threadIdx.x >> 5;
    const int lane = threadIdx.x & 31;
    const int hl   = lane >> 4;   // half-wave: selects K-pair 2*hl / 2*hl+1
    const int ln   = lane & 15;   // row (A) or column (B/D) index

    const int gid  = blockIdx.x * WAVES_PER_BLOCK + wave; // 0..NTILES*NSPLIT-1
    const int sp   = gid / NTILES;                        // key split id
    const int tile = gid - sp * NTILES;                   // query tile id
    const int b    = tile / QTILES;
    const int qt   = tile - b * QTILES;

    // split boundaries in units of 4-key-tile groups: 12/12/12/13
    const int gstart = (KGROUPS * sp) / NSPLIT;
    const int gend   = (KGROUPS * (sp + 1)) / NSPLIT;

    float* Lw = lds[wave];

    // ---- preload Q tile as A-operands, with the score scale (x8) folded in
    const float* Qb = Q + ((size_t)(b * NPIX + qt * 16 + ln)) * CH + 2 * hl;
    v2f qa[16];
#pragma unroll
    for (int c4 = 0; c4 < 16; ++c4) {
        v2f t = *(const v2f*)(Qb + 4 * c4);
        qa[c4] = t * 8.0f; // reference divides by d^-1/2 => multiply by 8
    }

    // ---- online-softmax state; acc D-layout row = j + 8*hl, col = t*16+ln
    float mrun[8], lrun[8];
    v8f acc[4];
#pragma unroll
    for (int j = 0; j < 8; ++j) { mrun[j] = -INFINITY; lrun[j] = 0.f; }
#pragma unroll
    for (int t = 0; t < 4; ++t)
#pragma unroll
        for (int j = 0; j < 8; ++j) acc[t][j] = 0.f;

    const float* Vbase = Vt + (size_t)b * CH * NPIX; // [c][n]

    for (int kb = gstart; kb < gend; ++kb) {
        // ---- S(4 tiles) = Q · K^T : 64 WMMAs, rotated over 4 accumulators
        const float* Kb = K + ((size_t)(b * NPIX + kb * 64 + ln)) * CH + 2 * hl;
        v8f s4[4];
#pragma unroll
        for (int u = 0; u < 4; ++u)
#pragma unroll
            for (int j = 0; j < 8; ++j) s4[u][j] = 0.f;
#pragma unroll
        for (int c4 = 0; c4 < 16; ++c4) {
#pragma unroll
            for (int u = 0; u < 4; ++u) {
                v2f kv = *(const v2f*)(Kb + (size_t)u * 16 * CH + 4 * c4);
                s4[u] = __builtin_amdgcn_wmma_f32_16x16x4_f32(
                            false, qa[c4], false, kv, (short)0, s4[u], false, false);
            }
        }

        // ---- online softmax over 64 keys: one DPP reduction pass per row
        float corr[8];
#pragma unroll
        for (int j = 0; j < 8; ++j) {
            float mt = fmaxf(fmaxf(s4[0][j], s4[1][j]),
                             fmaxf(s4[2][j], s4[3][j]));
            mt = redmax16(mt);
            float mn = fmaxf(mrun[j], mt);
            corr[j]  = __expf(mrun[j] - mn);
            mrun[j]  = mn;
#pragma unroll
            for (int u = 0; u < 4; ++u)
                s4[u][j] = __expf(s4[u][j] - mn); // P in-place
            float sum = (s4[0][j] + s4[1][j]) + (s4[2][j] + s4[3][j]);
            sum = redsum16(sum);
            lrun[j] = lrun[j] * corr[j] + sum;
        }
#pragma unroll
        for (int t = 0; t < 4; ++t)
#pragma unroll
            for (int j = 0; j < 8; ++j) acc[t][j] *= corr[j];

        // ---- bounce P (16x64) through LDS: D layout -> A layout (stride 66)
#pragma unroll
        for (int u = 0; u < 4; ++u)
#pragma unroll
            for (int j = 0; j < 8; ++j)
                Lw[(j + 8 * hl) * 66 + u * 16 + ln] = s4[u][j];

        // ---- ctx += P · V : contraction K=64, 64 WMMAs rotated over acc[0..3]
        const float* Vb = Vbase + kb * 64 + 2 * hl;
#pragma unroll
        for (int kk = 0; kk < 16; ++kk) {
            v2f pa = *(const v2f*)(&Lw[ln * 66 + kk * 4 + 2 * hl]);
#pragma unroll
            for (int t = 0; t < 4; ++t) {
                v2f vb = *(const v2f*)(Vb + (size_t)(t * 16 + ln) * NPIX + kk * 4);
                acc[t] = __builtin_amdgcn_wmma_f32_16x16x4_f32(
                             false, pa, false, vb, (short)0, acc[t], false, false);
            }
        }
    }

    // ---- write partials (unnormalized acc in D layout, plus m/l per row)
    float* pa = partAcc + ((size_t)(tile * NSPLIT + sp)) * (16 * 64);
#pragma unroll
    for (int t = 0; t < 4; ++t)
#pragma unroll
        for (int j = 0; j < 8; ++j)
            pa[(j + 8 * hl) * 64 + t * 16 + ln] = acc[t][j];

    float* pm = partML + ((size_t)(tile * NSPLIT + sp)) * 32;
    if (ln == 0) {
#pragma unroll
        for (int j = 0; j < 8; ++j) {
            pm[j + 8 * hl]      = mrun[j];
            pm[16 + j + 8 * hl] = lrun[j];
        }
    }
}

// -------------------------------------------------------------------------
// Kernel 3: merge the NSPLIT partials (softmax-weighted), then project by
// Wp and add bias.  One wave32 per query tile.
// -------------------------------------------------------------------------
__global__ __launch_bounds__(32 * WAVES_PER_BLOCK) void flash_combine_proj(
    const float* __restrict__ partAcc, const float* __restrict__ partML,
    const float* __restrict__ Wp, const float* __restrict__ bp,
    float* __restrict__ out)
{
    __shared__ float lds[WAVES_PER_BLOCK][1056];

    const int wave = threadIdx.x >> 5;
    const int lane = threadIdx.x & 31;
    const int hl   = lane >> 4;
    const int ln   = lane & 15;

    const int tile = blockIdx.x * WAVES_PER_BLOCK + wave; // 0..NTILES-1
    const int b    = tile / QTILES;
    const int qt   = tile - b * QTILES;

    float* Lw = lds[wave];

    // ---- per-row merge weights: w_s = exp(m_s - M), denom = sum l_s * w_s
    const float* pm = partML + (size_t)tile * NSPLIT * 32;
    float w[NSPLIT][8], invd[8];
#pragma unroll
    for (int j = 0; j < 8; ++j) {
        int row = j + 8 * hl;
        float m0 = pm[0 * 32 + row], m1 = pm[1 * 32 + row];
        float m2 = pm[2 * 32 + row], m3 = pm[3 * 32 + row];
        float M  = fmaxf(fmaxf(m0, m1), fmaxf(m2, m3));
        w[0][j] = __expf(m0 - M);
        w[1][j] = __expf(m1 - M);
        w[2][j] = __expf(m2 - M);
        w[3][j] = __expf(m3 - M);
        float denom = pm[0 * 32 + 16 + row] * w[0][j]
                    + pm[1 * 32 + 16 + row] * w[1][j]
                    + pm[2 * 32 + 16 + row] * w[2][j]
                    + pm[3 * 32 + 16 + row] * w[3][j];
        invd[j] = 1.0f / denom;
    }

    // ---- combined ctx in D layout
    const float* pa = partAcc + (size_t)tile * NSPLIT * (16 * 64);
    v8f acc[4];
#pragma unroll
    for (int t = 0; t < 4; ++t) {
#pragma unroll
        for (int j = 0; j < 8; ++j) {
            int idx = (j + 8 * hl) * 64 + t * 16 + ln;
            float v = pa[0 * 1024 + idx] * w[0][j]
                    + pa[1 * 1024 + idx] * w[1][j]
                    + pa[2 * 1024 + idx] * w[2][j]
                    + pa[3 * 1024 + idx] * w[3][j];
            acc[t][j] = v * invd[j];
        }
    }

    // ---- bounce ctx (16x64) through LDS: D layout -> A layout (stride 66)
#pragma unroll
    for (int t = 0; t < 4; ++t)
#pragma unroll
        for (int j = 0; j < 8; ++j)
            Lw[(j + 8 * hl) * 66 + t * 16 + ln] = acc[t][j];
    v2f ca[16];
#pragma unroll
    for (int c4 = 0; c4 < 16; ++c4)
        ca[c4] = *(const v2f*)(&Lw[ln * 66 + c4 * 4 + 2 * hl]);

    // ---- out = ctx @ Wp + bp : 64 WMMAs rotated over 4 output tiles
    v8f o[4];
#pragma unroll
    for (int t = 0; t < 4; ++t)
#pragma unroll
        for (int j = 0; j < 8; ++j) o[t][j] = 0.f;
#pragma unroll
    for (int c4 = 0; c4 < 16; ++c4) {
        int c0 = c4 * 4 + 2 * hl;
#pragma unroll
        for (int t = 0; t < 4; ++t) {
            v2f wb;
            wb.x = Wp[(c0 + 0) * ED + t * 16 + ln];
            wb.y = Wp[(c0 + 1) * ED + t * 16 + ln];
            o[t] = __builtin_amdgcn_wmma_f32_16x16x4_f32(
                       false, ca[c4], false, wb, (short)0, o[t], false, false);
        }
    }
#pragma unroll
    for (int t = 0; t < 4; ++t) {
        float bias = bp[t * 16 + ln];
#pragma unroll
        for (int j = 0; j < 8; ++j) {
            int row = qt * 16 + j + 8 * hl;
            out[((size_t)(b * NPIX + row)) * ED + t * 16 + ln] = o[t][j] + bias;
        }
    }
}

// -------------------------------------------------------------------------
extern "C" void kernel_launch(void* const* d_in, const int* in_sizes, int n_in,
                              void* d_out, int out_size, void* d_ws, size_t ws_size,
                              hipStream_t stream) {
    const float* x  = (const float*)d_in[0];
    const float* wq = (const float*)d_in[1];
    const float* bq = (const float*)d_in[2];
    const float* wk = (const float*)d_in[3];
    const float* bk = (const float*)d_in[4];
    const float* wv = (const float*)d_in[5];
    const float* bv = (const float*)d_in[6];
    const float* Wp = (const float*)d_in[7];
    const float* bp = (const float*)d_in[8];
    float* out = (float*)d_out;

    const size_t planeSz = (size_t)BATCH * NPIX * CH; // 1.6M floats
    float* q       = (float*)d_ws;
    float* k       = q + planeSz;
    float* vt      = k + planeSz;
    float* partAcc = vt + planeSz;                         // NTILES*NSPLIT*1024
    float* partML  = partAcc + (size_t)NTILES * NSPLIT * 1024; // NTILES*NSPLIT*32

    qkv_dwconv<<<BATCH * NPIX, 64, 0, stream>>>(x, wq, bq, wk, bk, wv, bv, q, k, vt);

    flash_attn_split<<<(NTILES * NSPLIT) / WAVES_PER_BLOCK,
                       32 * WAVES_PER_BLOCK, 0, stream>>>(q, k, vt, partAcc, partML);

    flash_combine_proj<<<NTILES / WAVES_PER_BLOCK,
                         32 * WAVES_PER_BLOCK, 0, stream>>>(partAcc, partML, Wp, bp, out);
}